// LGQuantizer_18648747999574
// MI455X (gfx1250) — compile-verified
//
#include <hip/hip_runtime.h>
#include <cstdint>

typedef float v2f __attribute__((ext_vector_type(2)));
typedef float v8f __attribute__((ext_vector_type(8)));

#define N_TOT 4096
#define CGRP  8
#define KCB   1024
#define DSUB  32
#define DFULL 256
#define INV_TEMP (1.0f/(2.0f + 1e-8f))

// workspace float layout
#define WS_CBSQ   0
#define WS_QSUM   8192
#define WS_COMMIT 16384
#define WS_QD     16385
#define WS_QLOGQ  16386

// d_out float offsets (outputs concatenated in return order)
#define OUT_QUANT 0
#define OUT_IDX   1048576
#define OUT_Q     1081344
#define OUT_SCAL  34635776

// ---------------- init: zero accumulators ----------------
__global__ void lgq_init(float* ws) {
    int i = blockIdx.x * blockDim.x + threadIdx.x;
    if (i < 8192 + 3) ws[WS_QSUM + i] = 0.0f;
}

// ---------------- codebook squared norms ----------------
__global__ void lgq_cbsq(const float* __restrict__ cb, float* __restrict__ ws) {
    int i = blockIdx.x * blockDim.x + threadIdx.x;
    if (i < CGRP * KCB) {
        const float* row = cb + (size_t)i * DSUB;
        float s = 0.f;
#pragma unroll
        for (int dd = 0; dd < DSUB; ++dd) s += row[dd] * row[dd];
        ws[WS_CBSQ + i] = s;
    }
}

// ---------------- main: WMMA cross + softmax + outputs ----------------
__global__ __launch_bounds__(128) void lgq_main(
    const float* __restrict__ z, const float* __restrict__ cb,
    float* __restrict__ ws, float* __restrict__ out_quant,
    float* __restrict__ out_idx, float* __restrict__ out_q) {

    extern __shared__ float sm[];
    float* sD      = sm;               // [16][1024] dists, later q
    float* sA      = sm + 16384;       // [16][32] zf tile
    float* sZsq    = sm + 16896;       // [16]
    float* sMin    = sm + 16912;       // [128]
    float* sS      = sm + 17040;       // [128]
    float* sT2     = sm + 17168;       // [128]
    float* sT3     = sm + 17296;       // [128]
    float* sRowMin = sm + 17424;       // [16]
    float* sRowRcp = sm + 17440;       // [16]
    float* sRed    = sm + 17456;       // [128]
    int*   sArg    = (int*)(sm + 17584);   // [128]
    int*   sRowArg = (int*)(sm + 17712);   // [16]

    const int t       = threadIdx.x;
    const int tile    = blockIdx.x;
    const int c       = tile & 7;
    const int rowtile = tile >> 3;
    const int nbase   = rowtile << 4;
    const int b       = nbase >> 10;     // H*W = 1024
    const int hwbase  = nbase & 1023;
    const float* cbc  = cb + (size_t)c * KCB * DSUB;
    const float* cbsq = ws + WS_CBSQ + c * KCB;

    // P0: stage zf tile (16 rows x 32 dims) into LDS (read-once -> NT load)
    for (int j = 0; j < 4; ++j) {
        int idx = t + 128 * j;
        int r = idx & 15, dd = idx >> 4;
        sA[r * DSUB + dd] = __builtin_nontemporal_load(
            z + ((size_t)(b * DFULL + c * DSUB + dd) << 10) + hwbase + r);
    }
    __syncthreads();
    if (t < 16) {
        float s = 0.f;
        for (int dd = 0; dd < DSUB; ++dd) { float v = sA[t * DSUB + dd]; s += v * v; }
        sZsq[t] = s;
    }
    __syncthreads();

    // P1: cross = A(16x32) @ B(32x16 per K-tile) via v_wmma_f32_16x16x4_f32
    const int wave = t >> 5;
    const int lane = t & 31;
    const int half = lane >> 4;      // K-pair selector (A/B f32 layout, ISA 7.12.2)
    const int mn   = lane & 15;      // A: row M ; B: col N
    float ar[16];
#pragma unroll
    for (int kk = 0; kk < 8; ++kk) {
        int dd0 = kk * 4 + 2 * half;
        ar[2 * kk]     = sA[mn * DSUB + dd0];
        ar[2 * kk + 1] = sA[mn * DSUB + dd0 + 1];
    }
    float zsqs[8];
#pragma unroll
    for (int i = 0; i < 8; ++i) zsqs[i] = sZsq[i + 8 * half];

    for (int kt = 0; kt < 16; ++kt) {        // each wave covers K range [wave*256, +256)
        const int col = wave * 256 + kt * 16 + mn;
        const float* crow = cbc + (size_t)col * DSUB;
        v8f acc = {0.f, 0.f, 0.f, 0.f, 0.f, 0.f, 0.f, 0.f};
#pragma unroll
        for (int kk = 0; kk < 8; ++kk) {
            int dd0 = kk * 4 + 2 * half;
            v2f av; av.x = ar[2 * kk]; av.y = ar[2 * kk + 1];
            v2f bv; bv.x = crow[dd0];  bv.y = crow[dd0 + 1];
            acc = __builtin_amdgcn_wmma_f32_16x16x4_f32(
                false, av, false, bv, (short)0, acc, false, false);
        }
        float cq = cbsq[col];
#pragma unroll
        for (int i = 0; i < 8; ++i) {
            int row = i + 8 * half;
            sD[row * KCB + col] = zsqs[i] - 2.0f * acc[i] + cq;
        }
    }
    __syncthreads();

    // P2: per-row min dist + argmin (min dist == max logit for softmax)
    {
        int r = t >> 3, seg = t & 7, kb = seg * 128;
        float dmin = 3.4e38f; int karg = 0;
        for (int i = 0; i < 128; ++i) {
            int k = kb + ((i + t) & 127);       // skew to spread LDS banks
            float d = sD[r * KCB + k];
            if (d < dmin || (d == dmin && k < karg)) { dmin = d; karg = k; }
        }
        sMin[t] = dmin; sArg[t] = karg;
    }
    __syncthreads();
    if (t < 16) {
        float dmin = sMin[t * 8]; int karg = sArg[t * 8];
        for (int s2 = 1; s2 < 8; ++s2) {
            float d = sMin[t * 8 + s2]; int k = sArg[t * 8 + s2];
            if (d < dmin || (d == dmin && k < karg)) { dmin = d; karg = k; }
        }
        sRowMin[t] = dmin; sRowArg[t] = karg;
        __builtin_nontemporal_store((float)karg,
            out_idx + ((b * CGRP + c) << 10) + hwbase + t);
    }
    __syncthreads();

    // P3: sum of exp, plus sum q~*logit and sum q~*dist for loss scalars
    {
        int r = t >> 3, seg = t & 7, kb = seg * 128;
        float dmin = sRowMin[r];
        float s = 0.f, t2 = 0.f, t3 = 0.f;
        for (int i = 0; i < 128; ++i) {
            int k = kb + ((i + t) & 127);
            float d  = sD[r * KCB + k];
            float lg = (dmin - d) * INV_TEMP;   // <= 0, no overflow
            float e  = __expf(lg);
            s += e; t2 += e * lg; t3 += e * d;
        }
        sS[t] = s; sT2[t] = t2; sT3[t] = t3;
    }
    __syncthreads();
    if (t < 16) {
        float S = 0.f, T2 = 0.f, T3 = 0.f;
        for (int s2 = 0; s2 < 8; ++s2) {
            S += sS[t * 8 + s2]; T2 += sT2[t * 8 + s2]; T3 += sT3[t * 8 + s2];
        }
        sRowRcp[t] = 1.0f / S;
        sRed[t]      = T3 / S;                 // sum_k q*dist  (this row)
        sRed[16 + t] = T2 / S - __logf(S);     // sum_k q*log q (this row)
    }
    __syncthreads();
    if (t == 0) {
        float a = 0.f, bb = 0.f;
        for (int r = 0; r < 16; ++r) { a += sRed[r]; bb += sRed[16 + r]; }
        atomicAdd(&ws[WS_QD], a);
        atomicAdd(&ws[WS_QLOGQ], bb);
    }

    // P4: normalized q -> global. 134 MB streaming write: non-temporal so the
    // L2 keeps cb/cbsq resident instead of caching the q stream.
    for (int j = 0; j < 8; ++j) {
        int k = j * 128 + t;
        float accq = 0.f;
        size_t qb = ((size_t)nbase * CGRP + c) * KCB + k;
        for (int r = 0; r < 16; ++r) {
            float d = sD[r * KCB + k];
            float q = __expf((sRowMin[r] - d) * INV_TEMP) * sRowRcp[r];
            sD[r * KCB + k] = q;                     // reuse LDS tile as q
            __builtin_nontemporal_store(q, out_q + qb + (size_t)r * CGRP * KCB);
            accq += q;
        }
        atomicAdd(&ws[WS_QSUM + c * KCB + k], accq);
    }
    __syncthreads();

    // P5: soft reconstruction z_q = q @ cb and commitment partial
    {
        int dd = t & 31, rg = t >> 5;
        float cpart = 0.f;
        for (int jr = 0; jr < 4; ++jr) {
            int r = rg * 4 + jr;
            float zq = 0.f;
            for (int k = 0; k < KCB; ++k)
                zq += sD[r * KCB + k] * cbc[(size_t)k * DSUB + dd];
            float diff = sA[r * DSUB + dd] - zq;
            cpart += diff * diff;
        }
        sRed[t] = cpart;
    }
    __syncthreads();
    if (t == 0) {
        float s = 0.f;
        for (int i = 0; i < 128; ++i) s += sRed[i];
        atomicAdd(&ws[WS_COMMIT], s);
    }

    // P6: hard-quantized output (straight-through forward value)
    for (int j = 0; j < 4; ++j) {
        int idx = t + 128 * j;
        int r = idx & 15, dd = idx >> 4;
        int ks = sRowArg[r];
        __builtin_nontemporal_store(cbc[(size_t)ks * DSUB + dd],
            out_quant + ((size_t)(b * DFULL + c * DSUB + dd) << 10) + hwbase + r);
    }
}

// ---------------- finalize scalars ----------------
__global__ __launch_bounds__(256) void lgq_finalize(const float* __restrict__ ws,
                                                    float* __restrict__ out) {
    __shared__ float red[256];
    int t = threadIdx.x;
    float bal = 0.f;
    for (int i = t; i < CGRP * KCB; i += 256) {
        float qbar = ws[WS_QSUM + i] * (1.0f / (float)N_TOT);
        bal += qbar * __logf(qbar * (float)KCB + 1e-8f);
    }
    red[t] = bal;
    __syncthreads();
    for (int s = 128; s > 0; s >>= 1) {
        if (t < s) red[t] += red[t + s];
        __syncthreads();
    }
    if (t == 0) {
        float balance = red[0] / (float)CGRP;
        float commit  = ws[WS_COMMIT] / (float)(N_TOT * CGRP * DSUB);
        float qd = ws[WS_QD], qlq = ws[WS_QLOGQ];
        float fe   = (qd * INV_TEMP + qlq) / (float)(N_TOT * CGRP) + __logf((float)KCB);
        float conf = -qlq / (float)(N_TOT * CGRP);
        out[OUT_SCAL + 0] = commit;
        out[OUT_SCAL + 1] = fe;
        out[OUT_SCAL + 2] = conf;
        out[OUT_SCAL + 3] = balance;
        out[OUT_SCAL + 4] = 1.0f;   // TAU
    }
}

extern "C" void kernel_launch(void* const* d_in, const int* in_sizes, int n_in,
                              void* d_out, int out_size, void* d_ws, size_t ws_size,
                              hipStream_t stream) {
    const float* z  = (const float*)d_in[0];
    const float* cb = (const float*)d_in[1];
    float* out = (float*)d_out;
    float* ws  = (float*)d_ws;

    lgq_init<<<(8192 + 3 + 255) / 256, 256, 0, stream>>>(ws);
    lgq_cbsq<<<(CGRP * KCB + 255) / 256, 256, 0, stream>>>(cb, ws);

    const size_t shmem = 17728u * sizeof(float);   // ~70 KB of the 320 KB WGP LDS
    lgq_main<<<2048, 128, shmem, stream>>>(z, cb, ws,
        out + OUT_QUANT, out + OUT_IDX, out + OUT_Q);

    lgq_finalize<<<1, 256, 0, stream>>>(ws, out);
}